// FeatureFuser_72533407695360
// MI455X (gfx1250) — compile-verified
//
#include <hip/hip_runtime.h>
#include <hip/hip_bf16.h>

// Problem constants from the reference
#define HH 512
#define WW 512
#define KK 4
#define CELL 128            // H / GRID_SIZE
#define WIN  384            // WINDOW_GRID_SIZE * CELL

typedef float v4f __attribute__((ext_vector_type(4)));

__global__ __launch_bounds__(256)
void FeatureFuser_72533407695360_kernel(const float* __restrict__ sampling,
                                        const float* __restrict__ refined,
                                        const int*   __restrict__ regions,
                                        float* __restrict__ out)
{
    const int b = blockIdx.y;

    // ---- Stage this batch's 8 region ints into LDS via the gfx1250 async
    // ---- global->LDS copy path (ASYNCcnt), then fence with s_wait_asynccnt.
    __shared__ __align__(16) int s_reg[2 * KK];
    if (threadIdx.x < 2 * KK) {
        const int* gp = regions + b * (2 * KK) + (int)threadIdx.x;
        // Generic pointers to LDS carry the LDS byte offset in the low 32 bits.
        unsigned lds_off = (unsigned)(unsigned long long)(&s_reg[threadIdx.x]);
        asm volatile("global_load_async_to_lds_b32 %0, %1, off"
                     :: "v"(lds_off), "v"(gp) : "memory");
    }
    asm volatile("s_wait_asynccnt 0" ::: "memory");
    __syncthreads();

    // ---- Window bounds per k (block-uniform values, cheap VALU/SALU)
    int y0[KK], y1[KK], x0[KK], x1[KK];
#pragma unroll
    for (int k = 0; k < KK; ++k) {
        int ry = s_reg[2 * k + 0];
        int rx = s_reg[2 * k + 1];
        int yy0 = min(max(ry * CELL, 0), HH);
        int xx0 = min(max(rx * CELL, 0), WW);
        y0[k] = yy0;  y1[k] = min(yy0 + WIN, HH);
        x0[k] = xx0;  x1[k] = min(xx0 + WIN, WW);
    }

    // ---- Each lane owns 4 consecutive pixels of one row
    const int p = (blockIdx.x * blockDim.x + threadIdx.x) * 4;
    const int y = p >> 9;          // p / 512
    const int x = p & (WW - 1);    // p % 512
    const long long img  = (long long)HH * WW;
    const long long base = (long long)b * img + p;

    // Winner = highest k whose window covers the pixel; -1 -> sampling map
    int sel0 = -1, sel1 = -1, sel2 = -1, sel3 = -1;
#pragma unroll
    for (int k = 0; k < KK; ++k) {
        if (y >= y0[k] && y < y1[k]) {
            if (x     >= x0[k] && x     < x1[k]) sel0 = k;
            if (x + 1 >= x0[k] && x + 1 < x1[k]) sel1 = k;
            if (x + 2 >= x0[k] && x + 2 < x1[k]) sel2 = k;
            if (x + 3 >= x0[k] && x + 3 < x1[k]) sel3 = k;
        }
    }

    // ---- Exactly one 128-bit non-temporal load per lane in the common case;
    // ---- scalar fallback only for groups straddling a window edge.
    v4f v;
    if ((sel0 == sel1) & (sel1 == sel2) & (sel2 == sel3)) {
        const float* src = (sel0 < 0)
            ? (sampling + base)
            : (refined + ((long long)b * KK + sel0) * img + p);
        v = __builtin_nontemporal_load((const v4f*)src);
    } else {
        const int s[4] = {sel0, sel1, sel2, sel3};
        float t[4];
#pragma unroll
        for (int j = 0; j < 4; ++j) {
            const float* src = (s[j] < 0)
                ? (sampling + base + j)
                : (refined + ((long long)b * KK + s[j]) * img + p + j);
            t[j] = __builtin_nontemporal_load(src);
        }
        v.x = t[0]; v.y = t[1]; v.z = t[2]; v.w = t[3];
    }

    // ---- sigmoid(x) = 1 / (1 + e^-x) via v_exp_f32 + v_rcp_f32
    v.x = __builtin_amdgcn_rcpf(1.0f + __expf(-v.x));
    v.y = __builtin_amdgcn_rcpf(1.0f + __expf(-v.y));
    v.z = __builtin_amdgcn_rcpf(1.0f + __expf(-v.z));
    v.w = __builtin_amdgcn_rcpf(1.0f + __expf(-v.w));

    __builtin_nontemporal_store(v, (v4f*)(out + base));
}

extern "C" void kernel_launch(void* const* d_in, const int* in_sizes, int n_in,
                              void* d_out, int out_size, void* d_ws, size_t ws_size,
                              hipStream_t stream) {
    const float* sampling = (const float*)d_in[0];   // (B,1,H,W) f32
    const float* refined  = (const float*)d_in[1];   // (B,K,1,H,W) f32
    const int*   regions  = (const int*)d_in[2];     // (B,K,2) int
    float* out = (float*)d_out;                      // (B,1,H,W) f32

    int B = (n_in >= 3) ? (in_sizes[2] / (2 * KK)) : 32;

    dim3 block(256);                                  // 8 wave32 waves
    dim3 grid((HH * WW) / (256 * 4), B);              // (256, 32)
    hipLaunchKernelGGL(FeatureFuser_72533407695360_kernel, grid, block, 0, stream,
                       sampling, refined, regions, out);
}